// Decoder_62783831933065
// MI455X (gfx1250) — compile-verified
//
#include <hip/hip_runtime.h>

typedef float v2f __attribute__((ext_vector_type(2)));
typedef float v8f __attribute__((ext_vector_type(8)));

#define B_      8
#define NCHUNK  63
#define CLEN    256
#define NFILT   512
#define TLEN    8192
#define FN      256     // filters per source
#define LDEC    16
#define SIGLEN  64000
#define TOKS    32      // tokens staged per block

// ---------------------------------------------------------------------------
// Kernel 1: fused chunk-OLA + mask*enc + per-source [16x256]x[256x16] GEMM
// via V_WMMA_F32_16X16X4_F32. One block = 64 threads = 2 waves, each wave
// owns a 16-token tile; 32-token masked tile staged in 64KB swizzled LDS.
// ---------------------------------------------------------------------------
__global__ __launch_bounds__(64) void fused_mask_gemm(
    const float* __restrict__ x,    // [B,63,256,512]
    const float* __restrict__ enc,  // [B,8192,512]
    const float* __restrict__ W,    // [2,256,16]
    float* __restrict__ dec)        // [2,B,8192,16]  (workspace)
{
  __shared__ float lds[TOKS * NFILT];          // 64 KB, XOR-swizzled (float4)

  const int tid   = threadIdx.x;               // 0..63
  const int gtok0 = blockIdx.x * TOKS;
  const int b     = gtok0 >> 13;               // / 8192
  const int t0    = gtok0 & (TLEN - 1);

  // ---- stage masked tile: 32 tok * 128 float4 = 4096 float4, 64 per thread
  const float4* x4 = (const float4*)x;
  const float4* e4 = (const float4*)enc;
  #pragma unroll 4
  for (int j = 0; j < 64; ++j) {
    int i4   = tid + j * 64;                   // 0..4095
    int tokL = i4 >> 7;                        // /128 -> local token
    int f4   = i4 & 127;                       // float4 index within 512 f
    int t    = t0 + tokL;
    int ch   = t >> 7;                         // hop = 128
    float mx = 0.f, my = 0.f, mz = 0.f, mw = 0.f;
    if (ch < NCHUNK) {                         // chunk ch, pos t-128ch in [0,128)
      float4 v = x4[(((size_t)b * NCHUNK + ch) * CLEN + (t - (ch << 7))) * 128 + f4];
      mx += v.x; my += v.y; mz += v.z; mw += v.w;
    }
    if (ch > 0) {                              // chunk ch-1, pos in [128,256)
      float4 v = x4[(((size_t)b * NCHUNK + (ch - 1)) * CLEN + (t - ((ch - 1) << 7))) * 128 + f4];
      mx += v.x; my += v.y; mz += v.z; mw += v.w;
    }
    float4 e = e4[((size_t)b * TLEN + t) * 128 + f4];
    float4 r; r.x = mx * e.x; r.y = my * e.y; r.z = mz * e.z; r.w = mw * e.w;
    int f4s = (f4 & ~15) | ((f4 ^ tokL) & 15); // bank swizzle, float4 granules
    *(float4*)&lds[tokL * NFILT + f4s * 4] = r;
  }
  __syncthreads();

  // ---- per-wave WMMA tile: M=16 tokens, N=16 taps, K=256 per source
  const int lane   = tid & 31;
  const int wave   = tid >> 5;
  const int laneN  = lane & 15;
  const int hi     = lane >> 4;                // half-wave: K pair select
  const int rowTok = wave * 16 + laneN;        // A-matrix row (M) for this lane
  const float* arow = &lds[rowTok * NFILT];

  v8f c0 = {0.f,0.f,0.f,0.f,0.f,0.f,0.f,0.f};  // source 0 accumulator
  v8f c1 = {0.f,0.f,0.f,0.f,0.f,0.f,0.f,0.f};  // source 1 accumulator

  #pragma unroll 8
  for (int kk = 0; kk < 64; ++kk) {            // K = 256 = 64 * 4
    // A fragments (fp32 16x4 layout: lanes<16 -> K=0,1 ; lanes>=16 -> K=2,3)
    int f4a = (kk & ~15) | ((kk ^ rowTok) & 15);              // src0: f4 = kk
    int kk1 = 64 + kk;                                        // src1: f4 = 64+kk
    int f4b = (kk1 & ~15) | ((kk1 ^ rowTok) & 15);
    v2f a0 = *(const v2f*)&arow[f4a * 4 + 2 * hi];
    v2f a1 = *(const v2f*)&arow[f4b * 4 + 2 * hi];
    // B fragments (4x16: VGPR0 -> K=2hi, VGPR1 -> K=2hi+1, N = laneN)
    int kr = 4 * kk + 2 * hi;
    v2f b0 = { W[(size_t)kr * LDEC + laneN],
               W[(size_t)(kr + 1) * LDEC + laneN] };
    v2f b1 = { W[(size_t)(FN + kr) * LDEC + laneN],
               W[(size_t)(FN + kr + 1) * LDEC + laneN] };
    c0 = __builtin_amdgcn_wmma_f32_16x16x4_f32(false, a0, false, b0,
                                               (short)0, c0, false, false);
    c1 = __builtin_amdgcn_wmma_f32_16x16x4_f32(false, a1, false, b1,
                                               (short)0, c1, false, false);
  }

  // ---- scatter D tile: VGPR r holds (M = r + 8*hi, N = laneN)
  const int tb = t0 + wave * 16 + 8 * hi;
  #pragma unroll
  for (int r = 0; r < 8; ++r) {
    int t = tb + r;
    dec[((size_t)b * TLEN + t) * LDEC + laneN]        = c0[r];  // n = 0
    dec[((size_t)(B_ + b) * TLEN + t) * LDEC + laneN] = c1[r];  // n = 1
  }
}

// ---------------------------------------------------------------------------
// Kernel 2: final overlap-add over (T, 16) with hop 8, slice to 64000.
// out[b,n,s] = dec[n,b,s/8, s&7] + dec[n,b,s/8-1, (s&7)+8]
// ---------------------------------------------------------------------------
__global__ __launch_bounds__(256) void ola_out(
    const float* __restrict__ dec,   // [2,B,8192,16]
    float* __restrict__ out)         // [B,2,64000]
{
  int idx = blockIdx.x * 256 + threadIdx.x;
  if (idx >= B_ * 2 * SIGLEN) return;
  int s  = idx % SIGLEN;
  int r  = idx / SIGLEN;
  int n  = r & 1;
  int b  = r >> 1;
  int t1 = s >> 3;
  int l1 = s & 7;
  const float* d = dec + (size_t)(n * B_ + b) * TLEN * LDEC;
  float v = d[(size_t)t1 * LDEC + l1];
  if (t1 > 0) v += d[(size_t)(t1 - 1) * LDEC + l1 + 8];
  out[((size_t)b * 2 + n) * SIGLEN + s] = v;
}

extern "C" void kernel_launch(void* const* d_in, const int* in_sizes, int n_in,
                              void* d_out, int out_size, void* d_ws, size_t ws_size,
                              hipStream_t stream) {
  const float* x   = (const float*)d_in[0];   // [8,63,256,512]
  const float* enc = (const float*)d_in[1];   // [8,8192,512]
  const float* W   = (const float*)d_in[2];   // [2,256,16]
  float* dec = (float*)d_ws;                  // 2*8*8192*16 f32 = 8 MB
  float* out = (float*)d_out;                 // [8,2,64000]

  (void)in_sizes; (void)n_in; (void)out_size; (void)ws_size;

  fused_mask_gemm<<<(B_ * TLEN) / TOKS, 64, 0, stream>>>(x, enc, W, dec);
  ola_out<<<(B_ * 2 * SIGLEN + 255) / 256, 256, 0, stream>>>(dec, out);
}